// MaxUnpooling2D_85839216377924
// MI455X (gfx1250) — compile-verified
//
#include <hip/hip_runtime.h>

// MaxUnpooling2D scatter-add for MI455X (gfx1250).
//
// Shapes (fixed by the reference):
//   updates [16, 64, 64, 128] f32, mask [16,64,64,128] i32 in [0, 2^21)
//   out     [16,128,128,128] f32, zero-init then out[b, m>>14, (m>>7)&127, c] += upd
//
// out flat index = (b << 21) | (m & 0x1FFF80) | channel
//
// Strategy: memory/atomic-bound op. Zero-fill output with B128 stores (stays
// L2-resident: 134MB output < 192MB L2), then scatter with nontemporal B128
// input loads (streamed once, don't thrash L2) and hardware non-returning
// fp32 global atomics (global_atomic_add_f32, RMW executes at L2).

typedef float v4f __attribute__((ext_vector_type(4)));
typedef int   v4i __attribute__((ext_vector_type(4)));

static constexpr int kC          = 128;
static constexpr int kBatchShift = 19;        // H*W*C   = 64*64*128 = 2^19
static constexpr int kSlabShift  = 21;        // OH*OW*C = 128*128*128 = 2^21
static constexpr int kYXMask     = 0x1FFF80;  // bits [20:7] of mask value

__global__ __launch_bounds__(256)
void zero_out_kernel(v4f* __restrict__ out4, int n4) {
    int i = blockIdx.x * 256 + threadIdx.x;
    if (i < n4) {
        out4[i] = (v4f)0.0f;   // global_store_b128, default RT -> L2 resident
    }
}

__global__ __launch_bounds__(256)
void scatter_add_kernel(const v4f* __restrict__ upd4,
                        const v4i* __restrict__ msk4,
                        float* __restrict__ out,
                        int n4) {
    int i = blockIdx.x * 256 + threadIdx.x;
    if (i >= n4) return;

    // Streamed-once inputs: nontemporal B128 loads so they don't evict the
    // L2-resident output lines the atomics are hammering.
    v4f u = __builtin_nontemporal_load(upd4 + i);
    v4i m = __builtin_nontemporal_load(msk4 + i);

    int base = i << 2;                              // flat element index
    int slab = (base >> kBatchShift) << kSlabShift; // b * OH*OW*C
    int f    = base & (kC - 1);                     // channel of lane 0 (mult of 4)

    // Non-returning hardware fp32 atomics (duplicate indices must sum).
    unsafeAtomicAdd(out + (slab + (m.x & kYXMask) + f + 0), u.x);
    unsafeAtomicAdd(out + (slab + (m.y & kYXMask) + f + 1), u.y);
    unsafeAtomicAdd(out + (slab + (m.z & kYXMask) + f + 2), u.z);
    unsafeAtomicAdd(out + (slab + (m.w & kYXMask) + f + 3), u.w);
}

extern "C" void kernel_launch(void* const* d_in, const int* in_sizes, int n_in,
                              void* d_out, int out_size, void* d_ws, size_t ws_size,
                              hipStream_t stream) {
    (void)n_in; (void)d_ws; (void)ws_size;

    const v4f* upd4 = (const v4f*)d_in[0];
    const v4i* msk4 = (const v4i*)d_in[1];
    float*     out  = (float*)d_out;

    // 1) zero the 134MB output (harness poisons it; we own initialization)
    int out4 = out_size >> 2;                       // 8,388,608 float4 stores
    int zb   = (out4 + 255) / 256;
    zero_out_kernel<<<zb, 256, 0, stream>>>((v4f*)out, out4);

    // 2) scatter-add, 4 channels per thread
    int in4 = in_sizes[0] >> 2;                     // 2,097,152 threads
    int sb  = (in4 + 255) / 256;
    scatter_add_kernel<<<sb, 256, 0, stream>>>(upd4, msk4, out, in4);
}